// HDHOGLayer_83708912599246
// MI455X (gfx1250) — compile-verified
//
#include <hip/hip_runtime.h>
#include <math.h>

// ---------------------------------------------------------------------------
// HOG (skimage-style) on MI455X / gfx1250.
//
// Roofline: HBM traffic = 33.5MB in + 39.9MB out -> ~3.2us floor @ 23.3TB/s.
// Compute ~0.5 GFLOP -> firmly memory-bound; WMMA not applicable.
// CDNA5-specific path used: GLOBAL_LOAD_ASYNC_TO_LDS (ASYNCcnt) staging of
// image row-bands into LDS, LDS ds_add_f32 histogram, L2-resident
// intermediates in d_ws.
// ---------------------------------------------------------------------------

#define ORI    9
#define CELLPX 8
#define IMG    512
#define NC     64      // cells per dimension  (512/8)
#define NB     62      // blocks per dimension (64-3+1)
#define BATCH  32

#if defined(__has_builtin)
#  if __has_builtin(__builtin_amdgcn_global_load_async_to_lds_b128)
#    define HAVE_ASYNC_LDS 1
#  endif
#  if __has_builtin(__builtin_amdgcn_s_wait_asynccnt)
#    define HAVE_WAIT_ASYNC_BUILTIN 1
#  endif
#endif

#ifdef HAVE_ASYNC_LDS
// Builtin prototype (from hipcc diagnostic): the async copy takes pointers to
// int __vector(4) in addrspace(1) (global) and addrspace(3) (LDS).
typedef int v4i_t __attribute__((ext_vector_type(4)));
typedef __attribute__((address_space(1))) v4i_t gv4i_t;   // global v4i
typedef __attribute__((address_space(3))) v4i_t lv4i_t;   // LDS v4i
#endif

// ---------------------------------------------------------------------------
// Kernel 1: per-cell orientation histogram.
// One workgroup per (batch image, cell-row). 256 threads = 8 wave32.
// Stages rows [r0-1, r0+8] (10 x 512 floats = 20KB) into LDS asynchronously,
// sqrt-transforms in place, computes gradients + hard binning, accumulates a
// 64-cell x 9-bin histogram with LDS float atomics, writes hist/64 to global.
// ---------------------------------------------------------------------------
__global__ __launch_bounds__(256)
void hog_cell_hist(const float* __restrict__ x, float* __restrict__ hist)
{
    __shared__ float s_img[10 * IMG];     // 20 KB staging tile
    __shared__ float s_hist[NC * ORI];    // 576 floats

    const int tid = threadIdx.x;
    const int wg  = blockIdx.x;           // b * 64 + cr
    const int b   = wg >> 6;
    const int cr  = wg & 63;
    const int r0  = cr * CELLPX;

    // Zero the shared histogram.
    for (int i = tid; i < NC * ORI; i += 256) s_hist[i] = 0.0f;

    // Zero-fill halo rows that fall outside the image (their g_row is masked
    // to zero anyway, but keep LDS contents defined for the sqrt pass).
    if (cr == 0)
        for (int c = tid; c < IMG; c += 256) s_img[c] = 0.0f;
    if (cr == NC - 1)
        for (int c = tid; c < IMG; c += 256) s_img[9 * IMG + c] = 0.0f;

    // Stage 10 rows x 512 floats = 1280 float4 (5 per thread) into LDS.
    const float* imgb = x + (size_t)b * IMG * IMG;
    for (int k = 0; k < 5; ++k) {
        const int f  = tid + k * 256;     // float4 index into the tile
        const int j  = f >> 7;            // tile row (128 float4 per row)
        const int c4 = f & 127;
        const int rr = r0 - 1 + j;        // image row
        if (rr >= 0 && rr < IMG) {
#ifdef HAVE_ASYNC_LDS
            // CDNA5 async global->LDS copy: 16B/lane, tracked by ASYNCcnt,
            // bypasses the VGPR file entirely.
            __builtin_amdgcn_global_load_async_to_lds_b128(
                (gv4i_t*)(imgb + (size_t)rr * IMG + c4 * 4),
                (lv4i_t*)(&s_img[j * IMG + c4 * 4]),
                0, 0);
#else
            ((float4*)s_img)[f] =
                ((const float4*)imgb)[rr * (IMG / 4) + c4];
#endif
        }
    }
#ifdef HAVE_ASYNC_LDS
#  ifdef HAVE_WAIT_ASYNC_BUILTIN
    __builtin_amdgcn_s_wait_asynccnt(0);
#  else
    asm volatile("s_wait_asynccnt 0" ::: "memory");
#  endif
#endif
    __syncthreads();

    // transform_sqrt=True: one sqrt per pixel (instead of 4-5 at gradient use).
    for (int i = tid; i < 10 * IMG; i += 256)
        s_img[i] = __builtin_sqrtf(s_img[i]);
    __syncthreads();

    // Gradients + hard binning + LDS-atomic histogram. 4096 px / 256 thr.
    const float RAD2DEG = 57.29577951308232f;   // 180/pi
    for (int k = 0; k < 16; ++k) {
        const int p = tid + k * 256;
        const int i = p >> 9;               // row within cell-row: 0..7
        const int c = p & 511;              // column
        const int r = r0 + i;               // image row

        const float up = s_img[i * IMG + c];
        const float dn = s_img[(i + 2) * IMG + c];
        const float gr = (r >= 1 && r <= IMG - 2) ? (dn - up) : 0.0f;

        const float lf = s_img[(i + 1) * IMG + (c > 0 ? c - 1 : 0)];
        const float rt = s_img[(i + 1) * IMG + (c < IMG - 1 ? c + 1 : c)];
        const float gc = (c >= 1 && c <= IMG - 2) ? (rt - lf) : 0.0f;

        const float mag = __builtin_sqrtf(gr * gr + gc * gc);
        if (mag > 0.0f) {
            float ang = atan2f(gr, gc) * RAD2DEG;   // (-180, 180]
            if (ang < 0.0f)    ang += 180.0f;       // -> [0, 180]
            if (ang >= 180.0f) ang -= 180.0f;       // -> [0, 180)
            int bin = (int)(ang * (1.0f / 20.0f));  // unit = 180/9
            bin = bin > (ORI - 1) ? (ORI - 1) : bin;
            // LDS float atomic add (ds_add_f32)
            __hip_atomic_fetch_add(&s_hist[(c >> 3) * ORI + bin], mag,
                                   __ATOMIC_RELAXED,
                                   __HIP_MEMORY_SCOPE_WORKGROUP);
        }
    }
    __syncthreads();

    // skimage divides each cell histogram by the cell area (8*8).
    float* hrow = hist + (size_t)wg * (NC * ORI);
    for (int i = tid; i < NC * ORI; i += 256)
        hrow[i] = s_hist[i] * (1.0f / 64.0f);
}

// ---------------------------------------------------------------------------
// Kernel 2: per-block inverse L2 norm.  hist is L2-resident (4.7MB << 192MB).
// ---------------------------------------------------------------------------
__global__ __launch_bounds__(256)
void hog_block_invnorm(const float* __restrict__ hist,
                       float* __restrict__ invn, int n)
{
    const int idx = blockIdx.x * 256 + threadIdx.x;
    if (idx >= n) return;
    const int b   = idx / (NB * NB);
    const int rem = idx - b * (NB * NB);
    const int br  = rem / NB;
    const int bc  = rem - br * NB;

    float s = 0.0f;
    #pragma unroll
    for (int r = 0; r < 3; ++r) {
        #pragma unroll
        for (int c = 0; c < 3; ++c) {
            const float* hp =
                hist + ((size_t)(b * NC + br + r) * NC + (bc + c)) * ORI;
            #pragma unroll
            for (int o = 0; o < ORI; ++o) { const float v = hp[o]; s += v * v; }
        }
    }
    // block_norm='L2': 1 / sqrt(sum + eps^2), eps = 1e-5
    invn[idx] = 1.0f / __builtin_sqrtf(s + 1e-10f);
}

// ---------------------------------------------------------------------------
// Kernel 3: normalized block features, skimage ravel order
//   flat = ((((b*62 + br)*62 + bc)*3 + r)*3 + c)*9 + o
// ---------------------------------------------------------------------------
__global__ __launch_bounds__(256)
void hog_write_blocks(const float* __restrict__ hist,
                      const float* __restrict__ invn,
                      float* __restrict__ out, int n)
{
    const int idx = blockIdx.x * 256 + threadIdx.x;
    if (idx >= n) return;
    int t = idx;
    const int o  = t % ORI; t /= ORI;
    const int c  = t % 3;   t /= 3;
    const int r  = t % 3;   t /= 3;
    const int bc = t % NB;  t /= NB;
    const int br = t % NB;  t /= NB;
    const int b  = t;

    const float h =
        hist[((size_t)(b * NC + br + r) * NC + (bc + c)) * ORI + o];
    out[idx] = h * invn[(b * NB + br) * NB + bc];
}

// ---------------------------------------------------------------------------
extern "C" void kernel_launch(void* const* d_in, const int* in_sizes, int n_in,
                              void* d_out, int out_size, void* d_ws,
                              size_t ws_size, hipStream_t stream)
{
    (void)in_sizes; (void)n_in; (void)ws_size;
    const float* x = (const float*)d_in[0];
    float* out  = (float*)d_out;

    // Workspace layout: hist [32][64][64][9] then invnorm [32][62][62].
    float* hist = (float*)d_ws;
    float* invn = hist + (size_t)BATCH * NC * NC * ORI;

    hog_cell_hist<<<BATCH * NC, 256, 0, stream>>>(x, hist);

    const int nblk = BATCH * NB * NB;
    hog_block_invnorm<<<(nblk + 255) / 256, 256, 0, stream>>>(hist, invn, nblk);

    hog_write_blocks<<<(out_size + 255) / 256, 256, 0, stream>>>(hist, invn,
                                                                 out, out_size);
}